// Classifier_36618891166177
// MI455X (gfx1250) — compile-verified
//
#include <hip/hip_runtime.h>
#include <hip/hip_bf16.h>

typedef float v2f __attribute__((ext_vector_type(2)));
typedef float v8f __attribute__((ext_vector_type(8)));

#define H 64
#define MT_PER_WAVE 4

// ---------------------------------------------------------------------------
// generic fill
__global__ void gcn_fill(float* __restrict__ p, float v, long n) {
    long i = (long)blockIdx.x * blockDim.x + threadIdx.x;
    long stride = (long)gridDim.x * blockDim.x;
    for (; i < n; i += stride) p[i] = v;
}

// ---------------------------------------------------------------------------
// degree counting: deg_out[src[e]] += 1 ; deg_in[dst[e]] += 1   (f32 exact)
__global__ void gcn_degree(const int* __restrict__ src, const int* __restrict__ dst,
                           float* __restrict__ deg_out, float* __restrict__ deg_in, int E) {
    int e = blockIdx.x * blockDim.x + threadIdx.x;
    if (e >= E) return;
    atomicAdd(&deg_out[src[e]], 1.0f);
    atomicAdd(&deg_in[dst[e]], 1.0f);
}

// in-place: x <- rsqrt(max(x, 1))
__global__ void gcn_norm(float* __restrict__ p, long n) {
    long i = (long)blockIdx.x * blockDim.x + threadIdx.x;
    if (i >= n) return;
    p[i] = rsqrtf(fmaxf(p[i], 1.0f));
}

// ---------------------------------------------------------------------------
// SpMM: agg[dst[e]][:] += norm_src[src[e]] * h[src[e]][:]
// 16 lanes per edge, float4 (16B) per lane -> global_load_b128 + 4x atomic_add_f32
__global__ void gcn_gather(const int* __restrict__ src, const int* __restrict__ dst,
                           const float* __restrict__ h, const float* __restrict__ norm_src,
                           float* __restrict__ agg, long n_work) {
    long t = (long)blockIdx.x * blockDim.x + threadIdx.x;
    if (t >= n_work) return;
    int e = (int)(t >> 4);
    int g = ((int)t & 15) * 4;
    int s = src[e];
    int d = dst[e];
    float ns = norm_src[s];
    const float4 v = *(const float4*)(h + (size_t)s * H + g);
    float* ap = agg + (size_t)d * H + g;
    atomicAdd(ap + 0, v.x * ns);
    atomicAdd(ap + 1, v.y * ns);
    atomicAdd(ap + 2, v.z * ns);
    atomicAdd(ap + 3, v.w * ns);
}

// ---------------------------------------------------------------------------
// out = relu(norm_dst[:,None] * (agg @ W) + b)  via V_WMMA_F32_16X16X4_F32
// One wave sweeps MT_PER_WAVE 16x16 output tiles in M, reusing the 16
// preloaded B fragments (entire 64x16 W column panel) across all of them.
// A-row addresses are clamped (not branch-guarded): an out-of-range row only
// feeds an output row that the store guard drops, so unconditional loads are
// safe and the K-loop is a clean load->wmma stream with full EXEC.
__global__ __launch_bounds__(32) void gcn_gemm_wmma(
        const float* __restrict__ agg,      // [n, 64]
        const float* __restrict__ W,        // [64, 64] row-major (k, n)
        const float* __restrict__ bias,     // [64]
        const float* __restrict__ norm_dst, // [n]
        float* __restrict__ out,            // [n, 64]
        int n_nodes) {
    const int ntiles_n = H / 16;                 // 4
    int tn     = blockIdx.x & (ntiles_n - 1);
    int tmBase = (blockIdx.x / ntiles_n) * MT_PER_WAVE;
    int lane = threadIdx.x;
    int half = lane >> 4;                        // 0: lanes 0-15, 1: lanes 16-31
    int l    = lane & 15;

    int colB = tn * 16 + l;                      // B: N = lane%16
    float bcol = bias[colB];

#if __has_builtin(__builtin_amdgcn_wmma_f32_16x16x4_f32)
    // ---- preload all 16 B fragments (64x16 panel of W) once per wave
    v2f bf[16];
    #pragma unroll
    for (int kk = 0; kk < 16; ++kk) {
        int ka = kk * 4 + 2 * half;              // this half-wave holds K=ka, ka+1
        bf[kk].x = W[ka * H + colB];
        bf[kk].y = W[(ka + 1) * H + colB];
    }

    #pragma unroll
    for (int m = 0; m < MT_PER_WAVE; ++m) {
        int tm   = tmBase + m;
        int rowA = tm * 16 + l;                  // A: M = lane%16
        // clamp (no branch): clamped rows only affect store-guarded outputs
        int rowSafe = rowA < n_nodes ? rowA : 0;
        const float* arow = agg + (size_t)rowSafe * H + 2 * half;

        v8f c = {};
        #pragma unroll
        for (int kk = 0; kk < 16; ++kk) {
            v2f a = *(const v2f*)(arow + kk * 4);   // 8B-aligned b64 load
            // (neg_a, A, neg_b, B, c_mod, C, reuse_a, reuse_b)
            c = __builtin_amdgcn_wmma_f32_16x16x4_f32(false, a, false, bf[kk],
                                                      (short)0, c, false, false);
        }

        // D layout: VGPR v -> (M = v + 8*half, N = lane&15)
        #pragma unroll
        for (int v = 0; v < 8; ++v) {
            int row = tm * 16 + v + 8 * half;
            if (row < n_nodes) {
                float val = c[v] * norm_dst[row] + bcol;
                out[(size_t)row * H + colB] = fmaxf(val, 0.0f);
            }
        }
    }
#else
    // scalar fallback (same math, no WMMA) -- only if builtin is absent
    for (int m = 0; m < MT_PER_WAVE; ++m) {
        int tm = tmBase + m;
        for (int v = 0; v < 8; ++v) {
            int row = tm * 16 + v + 8 * half;
            if (row < n_nodes) {
                float acc = 0.0f;
                for (int k = 0; k < H; ++k)
                    acc += agg[(size_t)row * H + k] * W[k * H + colB];
                float val = acc * norm_dst[row] + bcol;
                out[(size_t)row * H + colB] = fmaxf(val, 0.0f);
            }
        }
    }
#endif
}

// ---------------------------------------------------------------------------
// column-wise sum over nodes (for mean pooling); hsum must be pre-zeroed
__global__ void gcn_pool(const float* __restrict__ h, float* __restrict__ hsum, int n_nodes) {
    int col = threadIdx.x & (H - 1);
    int rowStart = (int)((blockIdx.x * blockDim.x + threadIdx.x) >> 6);
    int strideRows = (gridDim.x * blockDim.x) >> 6;
    float acc = 0.0f;
    for (int r = rowStart; r < n_nodes; r += strideRows)
        acc += h[(size_t)r * H + col];
    atomicAdd(&hsum[col], acc);
}

// hg = hsum / n ; out[10..73] = hg ; out[0..9] = hg @ Wc + bc
__global__ void gcn_head(const float* __restrict__ hsum, const float* __restrict__ Wc,
                         const float* __restrict__ bc, float* __restrict__ out, int n_nodes) {
    __shared__ float hg[H];
    int t = threadIdx.x;
    float v = hsum[t] / (float)n_nodes;
    hg[t] = v;
    out[10 + t] = v;
    __syncthreads();
    if (t < 10) {
        float acc = bc[t];
        #pragma unroll
        for (int k = 0; k < H; ++k) acc += hg[k] * Wc[k * 10 + t];
        out[t] = acc;
    }
}

// ---------------------------------------------------------------------------
extern "C" void kernel_launch(void* const* d_in, const int* in_sizes, int n_in,
                              void* d_out, int out_size, void* d_ws, size_t ws_size,
                              hipStream_t stream) {
    const float* features = (const float*)d_in[0];
    const int*   src      = (const int*)d_in[1];
    const int*   dst      = (const int*)d_in[2];
    const float* W1 = (const float*)d_in[3];  const float* b1 = (const float*)d_in[4];
    const float* W2 = (const float*)d_in[5];  const float* b2 = (const float*)d_in[6];
    const float* W3 = (const float*)d_in[7];  const float* b3 = (const float*)d_in[8];
    const float* Wc = (const float*)d_in[9];  const float* bc = (const float*)d_in[10];
    float* out = (float*)d_out;

    const int n = in_sizes[0] / H;       // 100000
    const int E = in_sizes[1];           // 1600000
    const size_t NH = (size_t)n * H;

    float* ws   = (float*)d_ws;
    float* agg  = ws;                    // [n, 64]
    float* bufA = ws + NH;               // [n, 64]
    float* bufB = ws + 2 * NH;           // [n, 64]
    float* nsrc = ws + 3 * NH;           // [n]
    float* ndst = nsrc + n;              // [n]  (contiguous with nsrc)
    float* hsum = ndst + n;              // [64] (contiguous with ndst)

    const int TB = 256;

    // zero norm_src, norm_dst, hsum in one shot (contiguous): 2n + 64 floats
    {
        long cnt = 2L * n + H;
        int blocks = (int)((cnt + TB - 1) / TB);
        gcn_fill<<<blocks, TB, 0, stream>>>(nsrc, 0.0f, cnt);
    }
    gcn_degree<<<(E + TB - 1) / TB, TB, 0, stream>>>(src, dst, nsrc, ndst, E);
    gcn_norm<<<(int)((2L * n + TB - 1) / TB), TB, 0, stream>>>(nsrc, 2L * n);

    const long gwork = (long)E * 16;
    const int gather_blocks = (int)((gwork + TB - 1) / TB);
    const int fill_blocks   = (int)((NH + TB - 1) / TB);
    const int mtiles        = (n + 15) / 16;
    const int gemm_blocks   = ((mtiles + MT_PER_WAVE - 1) / MT_PER_WAVE) * (H / 16);

    // ---- layer 1: features -> bufA
    gcn_fill<<<fill_blocks, TB, 0, stream>>>(agg, 0.0f, (long)NH);
    gcn_gather<<<gather_blocks, TB, 0, stream>>>(src, dst, features, nsrc, agg, gwork);
    gcn_gemm_wmma<<<gemm_blocks, 32, 0, stream>>>(agg, W1, b1, ndst, bufA, n);

    // ---- layer 2: bufA -> bufB
    gcn_fill<<<fill_blocks, TB, 0, stream>>>(agg, 0.0f, (long)NH);
    gcn_gather<<<gather_blocks, TB, 0, stream>>>(src, dst, bufA, nsrc, agg, gwork);
    gcn_gemm_wmma<<<gemm_blocks, 32, 0, stream>>>(agg, W2, b2, ndst, bufB, n);

    // ---- layer 3: bufB -> bufA
    gcn_fill<<<fill_blocks, TB, 0, stream>>>(agg, 0.0f, (long)NH);
    gcn_gather<<<gather_blocks, TB, 0, stream>>>(src, dst, bufB, nsrc, agg, gwork);
    gcn_gemm_wmma<<<gemm_blocks, 32, 0, stream>>>(agg, W3, b3, ndst, bufA, n);

    // ---- mean pool + classifier
    gcn_pool<<<256, TB, 0, stream>>>(bufA, hsum, n);
    gcn_head<<<1, H, 0, stream>>>(hsum, Wc, bc, out, n);
}